// RNNPoolOP_82334523064945
// MI455X (gfx1250) — compile-verified
//
#include <hip/hip_runtime.h>

// MI455X / gfx1250, wave32 only.
typedef __attribute__((ext_vector_type(2))) float v2f;
typedef __attribute__((ext_vector_type(8))) float v8f;

__device__ __forceinline__ v8f wmma4(v2f a, v2f b, v8f c) {
  // D(16x16,f32) = A(16x4,f32) * B(4x16,f32) + C
  return __builtin_amdgcn_wmma_f32_16x16x4_f32(
      /*neg_a=*/false, a, /*neg_b=*/false, b,
      /*c_mod=*/(short)0, c, /*reuse_a=*/false, /*reuse_b=*/false);
}

__device__ __forceinline__ float frcp(float x) { return __builtin_amdgcn_rcpf(x); }
__device__ __forceinline__ float sigm(float x) { return frcp(1.0f + __expf(-x)); }
__device__ __forceinline__ float tanh_fast(float x) {
  // tanh(x) = 2*sigmoid(2x) - 1 : one v_exp_f32 + one v_rcp_f32
  return 2.0f * frcp(1.0f + __expf(-2.0f * x)) - 1.0f;
}

// ---------------------------------------------------------------------------
// rnn1: 131072 sequences (row region [0,65536), col region [65536,131072)),
// each length 8, D=64 -> H=32.  One wave handles 16 sequences (M-tile).
// Writes h_final to scratch h1out[seq][32].
//
// Weight LDS image is pair-interleaved: sWb[p*32+c] = {W[2p][c], W[2p+1][c]}
// so a WMMA B-fragment (rows kb,kb+1 at column c; kb always even) is one
// aligned ds_load_b64 with no re-pairing movs.
// ---------------------------------------------------------------------------
__global__ __launch_bounds__(256) void rnn1_kernel(
    const float* __restrict__ x,    // [8192,8,8,64]
    const float* __restrict__ W1,   // [64,32]
    const float* __restrict__ U1,   // [32,32]
    const float* __restrict__ bgp,  // [32]
    const float* __restrict__ bhp,  // [32]
    const float* __restrict__ zeta,
    const float* __restrict__ nu,
    float* __restrict__ h1out) {    // [131072,32]
  __shared__ v2f  sWb[32 * 32];     // W1 pair-interleaved (p=0..31)
  __shared__ v2f  sUb[16 * 32];     // U1 pair-interleaved (p=0..15)
  __shared__ float sbg[32];
  __shared__ float sbh[32];
  __shared__ float hst[8][16][34];  // per-wave D->A staging, padded stride

  const int tid = threadIdx.x;
  for (int i = tid; i < 1024; i += 256) {
    const int p = i >> 5, c = i & 31;
    v2f t = {W1[p * 64 + c], W1[p * 64 + 32 + c]};
    sWb[i] = t;
  }
  for (int i = tid; i < 512; i += 256) {
    const int p = i >> 5, c = i & 31;
    v2f t = {U1[p * 64 + c], U1[p * 64 + 32 + c]};
    sUb[i] = t;
  }
  if (tid < 32) {
    sbg[tid] = bgp[tid];
    sbh[tid] = bhp[tid];
  }
  __syncthreads();

  const float zs = sigm(zeta[0]);
  const float ns = sigm(nu[0]);

  const int wave = tid >> 5;
  const int lane = tid & 31;
  const int tile = blockIdx.x * 8 + wave;  // 0..8191
  const int m    = lane & 15;              // A-row within M-tile / D-col
  const int kh   = lane >> 4;              // selects K pair within chunk
  const int s    = tile * 16 + m;          // this lane's sequence id

  // Per-lane global base (element offset) and per-timestep stride.
  int base, stride;
  if (s < 65536) {           // row sweep: x[b][r][t][:], seq = b*8+r
    base = s * 512;          // (s*8 + t)*64
    stride = 64;
  } else {                   // col sweep: x[b][t][c][:], seq' = b*8+c
    const int sp = s - 65536;
    base = ((sp >> 3) * 64 + (sp & 7)) * 64;  // (b*64 + t*8 + c)*64
    stride = 512;
  }

  const int col = m;
  const float bg0 = sbg[col], bg1v = sbg[col + 16];
  const float bh0 = sbh[col], bh1v = sbh[col + 16];

  float(*hs)[34] = hst[wave];

  v8f h0 = {};  // units 0..15, D-layout
  v8f h1 = {};  // units 16..31, D-layout
  const int mrow = 8 * kh;

  // ---- t = 0 (h == 0): pre = x0 @ W1 ----
  {
    const float* xp = x + base;
    v2f xa[16];
#pragma unroll
    for (int k = 0; k < 16; ++k) xa[k] = *(const v2f*)(xp + 4 * k + 2 * kh);
    v8f a0 = {};
    v8f a1 = {};
#pragma unroll
    for (int k = 0; k < 16; ++k) {
      const int p = 2 * k + kh;
      a0 = wmma4(xa[k], sWb[p * 32 + col], a0);
      a1 = wmma4(xa[k], sWb[p * 32 + col + 16], a1);
    }
#pragma unroll
    for (int v = 0; v < 8; ++v) {
      const float p0 = a0[v], p1 = a1[v];
      const float z0 = sigm(p0 + bg0);
      const float z1 = sigm(p1 + bg1v);
      const float c0 = tanh_fast(p0 + bh0);
      const float c1 = tanh_fast(p1 + bh1v);
      h0[v] = (zs * (1.0f - z0) + ns) * c0;
      h1[v] = (zs * (1.0f - z1) + ns) * c1;
    }
#pragma unroll
    for (int v = 0; v < 8; ++v) {
      hs[v + mrow][col]      = h0[v];
      hs[v + mrow][col + 16] = h1[v];
    }
    asm volatile("s_wait_dscnt 0x0" ::: "memory");  // drain LDS stores
  }

  // ---- t = 1..7: pre = x_t @ W1 + h @ U1 ----
#pragma unroll
  for (int t = 1; t < 8; ++t) {
    const float* xp = x + base + t * stride;
    v2f xa[16];
#pragma unroll
    for (int k = 0; k < 16; ++k) xa[k] = *(const v2f*)(xp + 4 * k + 2 * kh);
    v2f ha[8];
#pragma unroll
    for (int k = 0; k < 8; ++k) ha[k] = *(const v2f*)(&hs[m][4 * k + 2 * kh]);

    v8f a0 = {};
    v8f a1 = {};
#pragma unroll
    for (int k = 0; k < 16; ++k) {
      const int p = 2 * k + kh;
      a0 = wmma4(xa[k], sWb[p * 32 + col], a0);
      a1 = wmma4(xa[k], sWb[p * 32 + col + 16], a1);
    }
#pragma unroll
    for (int k = 0; k < 8; ++k) {
      const int p = 2 * k + kh;
      a0 = wmma4(ha[k], sUb[p * 32 + col], a0);
      a1 = wmma4(ha[k], sUb[p * 32 + col + 16], a1);
    }
#pragma unroll
    for (int v = 0; v < 8; ++v) {
      const float p0 = a0[v], p1 = a1[v];
      const float z0 = sigm(p0 + bg0);
      const float z1 = sigm(p1 + bg1v);
      const float c0 = tanh_fast(p0 + bh0);
      const float c1 = tanh_fast(p1 + bh1v);
      h0[v] = z0 * h0[v] + (zs * (1.0f - z0) + ns) * c0;
      h1[v] = z1 * h1[v] + (zs * (1.0f - z1) + ns) * c1;
    }
    if (t < 7) {
#pragma unroll
      for (int v = 0; v < 8; ++v) {
        hs[v + mrow][col]      = h0[v];
        hs[v + mrow][col + 16] = h1[v];
      }
      asm volatile("s_wait_dscnt 0x0" ::: "memory");
    }
  }

  // Emit final hidden state to scratch: h1out[seq][unit]
  // (per v: the wave covers full 128B rows -> coalesced 64B segments)
#pragma unroll
  for (int v = 0; v < 8; ++v) {
    const int r = (tile * 16 + mrow + v) * 32;
    h1out[r + col]      = h0[v];
    h1out[r + col + 16] = h1[v];
  }
}

// ---------------------------------------------------------------------------
// rnn2: 4 combos (row-fwd, row-rev, col-fwd, col-rev) x 8192 batch, length 8,
// H=32 -> 32.  One wave handles 16 batch items for one combo.
// out[b][combo*32 + u].
// ---------------------------------------------------------------------------
__global__ __launch_bounds__(256) void rnn2_kernel(
    const float* __restrict__ h1,   // [131072,32]
    const float* __restrict__ W2,   // [32,32]
    const float* __restrict__ U2,   // [32,32]
    const float* __restrict__ bgp,  // [32]
    const float* __restrict__ bhp,  // [32]
    const float* __restrict__ zeta,
    const float* __restrict__ nu,
    float* __restrict__ out) {      // [8192,128]
  __shared__ v2f  sWb[16 * 32];
  __shared__ v2f  sUb[16 * 32];
  __shared__ float sbg[32];
  __shared__ float sbh[32];
  __shared__ float hst[8][16][34];

  const int tid = threadIdx.x;
  for (int i = tid; i < 512; i += 256) {
    const int p = i >> 5, c = i & 31;
    v2f tw = {W2[p * 64 + c], W2[p * 64 + 32 + c]};
    v2f tu = {U2[p * 64 + c], U2[p * 64 + 32 + c]};
    sWb[i] = tw;
    sUb[i] = tu;
  }
  if (tid < 32) {
    sbg[tid] = bgp[tid];
    sbh[tid] = bhp[tid];
  }
  __syncthreads();

  const float zs = sigm(zeta[0]);
  const float ns = sigm(nu[0]);

  const int wave  = tid >> 5;
  const int lane  = tid & 31;
  const int wtile = blockIdx.x * 8 + wave;  // 0..2047
  const int combo = wtile & 3;              // 0:row-f 1:row-r 2:col-f 3:col-r
  const int bt    = wtile >> 2;             // 0..511
  const int m     = lane & 15;
  const int kh    = lane >> 4;
  const int b     = bt * 16 + m;

  const int srcbase = (combo >= 2) ? (65536 * 32) : 0;
  const int rev     = combo & 1;

  const int col = m;
  const float bg0 = sbg[col], bg1v = sbg[col + 16];
  const float bh0 = sbh[col], bh1v = sbh[col + 16];

  float(*hs)[34] = hst[wave];

  v8f h0  = {};
  v8f h1t = {};
  const int mrow = 8 * kh;

  // ---- t = 0 ----
  {
    const int tt = rev ? 7 : 0;
    const float* xp = h1 + srcbase + (b * 8 + tt) * 32;
    v2f xa[8];
#pragma unroll
    for (int k = 0; k < 8; ++k) xa[k] = *(const v2f*)(xp + 4 * k + 2 * kh);
    v8f a0 = {};
    v8f a1 = {};
#pragma unroll
    for (int k = 0; k < 8; ++k) {
      const int p = 2 * k + kh;
      a0 = wmma4(xa[k], sWb[p * 32 + col], a0);
      a1 = wmma4(xa[k], sWb[p * 32 + col + 16], a1);
    }
#pragma unroll
    for (int v = 0; v < 8; ++v) {
      const float p0 = a0[v], p1 = a1[v];
      const float z0 = sigm(p0 + bg0);
      const float z1 = sigm(p1 + bg1v);
      const float c0 = tanh_fast(p0 + bh0);
      const float c1 = tanh_fast(p1 + bh1v);
      h0[v]  = (zs * (1.0f - z0) + ns) * c0;
      h1t[v] = (zs * (1.0f - z1) + ns) * c1;
    }
#pragma unroll
    for (int v = 0; v < 8; ++v) {
      hs[v + mrow][col]      = h0[v];
      hs[v + mrow][col + 16] = h1t[v];
    }
    asm volatile("s_wait_dscnt 0x0" ::: "memory");
  }

  // ---- t = 1..7 ----
#pragma unroll
  for (int t = 1; t < 8; ++t) {
    const int tt = rev ? (7 - t) : t;
    const float* xp = h1 + srcbase + (b * 8 + tt) * 32;
    v2f xa[8];
#pragma unroll
    for (int k = 0; k < 8; ++k) xa[k] = *(const v2f*)(xp + 4 * k + 2 * kh);
    v2f ha[8];
#pragma unroll
    for (int k = 0; k < 8; ++k) ha[k] = *(const v2f*)(&hs[m][4 * k + 2 * kh]);

    v8f a0 = {};
    v8f a1 = {};
#pragma unroll
    for (int k = 0; k < 8; ++k) {
      const int p = 2 * k + kh;
      a0 = wmma4(xa[k], sWb[p * 32 + col], a0);
      a1 = wmma4(xa[k], sWb[p * 32 + col + 16], a1);
    }
#pragma unroll
    for (int k = 0; k < 8; ++k) {
      const int p = 2 * k + kh;
      a0 = wmma4(ha[k], sUb[p * 32 + col], a0);
      a1 = wmma4(ha[k], sUb[p * 32 + col + 16], a1);
    }
#pragma unroll
    for (int v = 0; v < 8; ++v) {
      const float p0 = a0[v], p1 = a1[v];
      const float z0 = sigm(p0 + bg0);
      const float z1 = sigm(p1 + bg1v);
      const float c0 = tanh_fast(p0 + bh0);
      const float c1 = tanh_fast(p1 + bh1v);
      h0[v]  = z0 * h0[v]  + (zs * (1.0f - z0) + ns) * c0;
      h1t[v] = z1 * h1t[v] + (zs * (1.0f - z1) + ns) * c1;
    }
    if (t < 7) {
#pragma unroll
      for (int v = 0; v < 8; ++v) {
        hs[v + mrow][col]      = h0[v];
        hs[v + mrow][col + 16] = h1t[v];
      }
      asm volatile("s_wait_dscnt 0x0" ::: "memory");
    }
  }

  // out[b][combo*32 + u]
#pragma unroll
  for (int v = 0; v < 8; ++v) {
    const int r = (bt * 16 + mrow + v) * 128 + combo * 32;
    out[r + col]      = h0[v];
    out[r + col + 16] = h1t[v];
  }
}

extern "C" void kernel_launch(void* const* d_in, const int* in_sizes, int n_in,
                              void* d_out, int out_size, void* d_ws, size_t ws_size,
                              hipStream_t stream) {
  (void)in_sizes; (void)n_in; (void)out_size; (void)ws_size;
  const float* x   = (const float*)d_in[0];
  const float* W1  = (const float*)d_in[1];
  const float* U1  = (const float*)d_in[2];
  const float* bg1 = (const float*)d_in[3];
  const float* bh1 = (const float*)d_in[4];
  const float* z1  = (const float*)d_in[5];
  const float* n1  = (const float*)d_in[6];
  const float* W2  = (const float*)d_in[7];
  const float* U2  = (const float*)d_in[8];
  const float* bg2 = (const float*)d_in[9];
  const float* bh2 = (const float*)d_in[10];
  const float* z2  = (const float*)d_in[11];
  const float* n2  = (const float*)d_in[12];
  float* out = (float*)d_out;
  float* h1s = (float*)d_ws;  // 131072*32 floats = 16 MiB scratch

  // 8192 sequence-tiles (16 seqs each), 8 waves per 256-thread block.
  rnn1_kernel<<<1024, 256, 0, stream>>>(x, W1, U1, bg1, bh1, z1, n1, h1s);
  // 2048 wave-tiles (4 combos x 512 batch-tiles), 8 waves per block.
  rnn2_kernel<<<256, 256, 0, stream>>>(h1s, W2, U2, bg2, bh2, z2, n2, out);
}